// MambaUpBlock_70317204570777
// MI455X (gfx1250) — compile-verified
//
#include <hip/hip_runtime.h>
#include <hip/hip_bf16.h>
#include <math.h>

typedef float f32x2 __attribute__((ext_vector_type(2)));
typedef float f32x8 __attribute__((ext_vector_type(8)));

static inline long cdivl(long a, long b) { return (a + b - 1) / b; }

// ---------------------------------------------------------------------------
// Generic fp32 WMMA GEMM:  C[m,n] = sum_k A[m,k]*B[k,n]  (+ optional epilogue)
// Stride-parameterized; one wave32 per 16x16 tile, V_WMMA_F32_16X16X4_F32.
// AK1/BK1: compile-time "K stride == 1" -> aligned f32x2 (b64) fragment loads.
// Fast path (wave-uniform interior tile): zero guards in the loop.
// Edge path: clamped indices + mask-multiply (branch-free loads).
// ACT: 0 = none, 1 = softplus(x + bias[n])
// ---------------------------------------------------------------------------
template <int ACT, int AK1, int BK1>
__launch_bounds__(256)
__global__ void gemm_wmma_f32(const float* __restrict__ A, const float* __restrict__ B,
                              float* __restrict__ C, const float* __restrict__ bias,
                              int M, int N, int K,
                              long am, long ak, long bk, long bn, long cm, long cn) {
  int wave = threadIdx.x >> 5;
  int lane = threadIdx.x & 31;
  int tilesN = (N + 15) >> 4;
  long tileId = (long)blockIdx.x * 8 + wave;
  int tM = (int)(tileId / tilesN);
  int tN = (int)(tileId % tilesN);
  if (tM * 16 >= M) return;  // wave-uniform exit
  int m0 = tM * 16, n0 = tN * 16;
  int lr = lane & 15;          // M (for A) / N (for B/C) index within tile
  int kk = (lane >> 4) << 1;   // K pair base: lanes 16..31 hold K+2,K+3

  f32x8 acc = {0.f, 0.f, 0.f, 0.f, 0.f, 0.f, 0.f, 0.f};
  bool interior = (m0 + 16 <= M) && (n0 + 16 <= N);  // wave-uniform

  if (interior) {
    const float* Ap = A + (long)(m0 + lr) * am + (long)kk * ak;
    const float* Bp = B + (long)(n0 + lr) * bn + (long)kk * bk;
    int k = 0;
    for (; k + 4 <= K; k += 4) {
      f32x2 av, bv;
      if (AK1) av = *(const f32x2*)Ap;            // aligned b64 load
      else     { av[0] = Ap[0]; av[1] = Ap[ak]; }
      if (BK1) bv = *(const f32x2*)Bp;            // aligned b64 load
      else     { bv[0] = Bp[0]; bv[1] = Bp[bk]; }
      acc = __builtin_amdgcn_wmma_f32_16x16x4_f32(false, av, false, bv,
                                                  (short)0, acc, false, false);
      Ap += 4 * ak;
      Bp += 4 * bk;
    }
    if (k < K) {  // K remainder: clamp + mask (no exec-divergent loads)
      int ka = k + kk, kb = ka + 1;
      float f0 = (ka < K) ? 1.f : 0.f;
      float f1 = (kb < K) ? 1.f : 0.f;
      long kac = (ka < K) ? ka : (K - 1);
      long kbc = (kb < K) ? kb : (K - 1);
      const float* Ar = A + (long)(m0 + lr) * am;
      const float* Br = B + (long)(n0 + lr) * bn;
      f32x2 av = {Ar[kac * ak] * f0, Ar[kbc * ak] * f1};
      f32x2 bv = {Br[kac * bk] * f0, Br[kbc * bk] * f1};
      acc = __builtin_amdgcn_wmma_f32_16x16x4_f32(false, av, false, bv,
                                                  (short)0, acc, false, false);
    }
    int mh = (lane >> 4) << 3;
    long crow = (long)(n0 + lr) * cn;
    float bb = (ACT == 1) ? bias[n0 + lr] : 0.f;
    for (int r = 0; r < 8; ++r) {
      float v = acc[r];
      if (ACT == 1) {
        v += bb;
        v = (v > 20.f) ? v : log1pf(__expf(v));
      }
      C[(long)(m0 + mh + r) * cm + crow] = v;
    }
  } else {
    int ma = m0 + lr, nb = n0 + lr;
    float mf = (ma < M) ? 1.f : 0.f;
    float nf = (nb < N) ? 1.f : 0.f;
    long mac = (ma < M) ? ma : (M - 1);
    long nbc = (nb < N) ? nb : (N - 1);
    const float* Ar = A + mac * am;
    const float* Br = B + nbc * bn;
    for (int k = 0; k < K; k += 4) {
      int ka = k + kk, kb = ka + 1;
      float f0 = (ka < K) ? mf : 0.f;
      float g0 = (ka < K) ? nf : 0.f;
      float f1 = (kb < K) ? mf : 0.f;
      float g1 = (kb < K) ? nf : 0.f;
      long kac = (ka < K) ? ka : (K - 1);
      long kbc = (kb < K) ? kb : (K - 1);
      f32x2 av = {Ar[kac * ak] * f0, Ar[kbc * ak] * f1};
      f32x2 bv = {Br[kac * bk] * g0, Br[kbc * bk] * g1};
      acc = __builtin_amdgcn_wmma_f32_16x16x4_f32(false, av, false, bv,
                                                  (short)0, acc, false, false);
    }
    int mh = (lane >> 4) << 3;
    for (int r = 0; r < 8; ++r) {
      int m = m0 + mh + r, n = n0 + lr;
      if (m < M && n < N) {
        float v = acc[r];
        if (ACT == 1) {
          v += bias[n];
          v = (v > 20.f) ? v : log1pf(__expf(v));
        }
        C[(long)m * cm + (long)n * cn] = v;
      }
    }
  }
}

// ---------------------------------------------------------------------------
// Elementwise / small kernels
// ---------------------------------------------------------------------------
__global__ void concat_nchw_to_nhwc(const float* __restrict__ x, const float* __restrict__ p,
                                    float* __restrict__ out, int Bsz, int H, int W) {
  long total = (long)Bsz * H * W * 96;
  long i = (long)blockIdx.x * blockDim.x + threadIdx.x;
  if (i >= total) return;
  int c = (int)(i % 96); long t = i / 96;
  int w = (int)(t % W); t /= W;
  int h = (int)(t % H); int b = (int)(t / H);
  float v;
  if (c < 64) v = x[(((long)b * 64 + c) * H + h) * W + w];
  else        v = p[(((long)b * 32 + (c - 64)) * H + h) * W + w];
  out[i] = v;
}

__global__ void layernorm_rows(const float* __restrict__ x, const float* __restrict__ g,
                               const float* __restrict__ bta, float* __restrict__ out,
                               long M, int C) {
  long m = (long)blockIdx.x * blockDim.x + threadIdx.x;
  if (m >= M) return;
  const float* xr = x + m * C;
  float mu = 0.f;
  for (int c = 0; c < C; ++c) mu += xr[c];
  mu /= (float)C;
  float var = 0.f;
  for (int c = 0; c < C; ++c) { float d = xr[c] - mu; var += d * d; }
  var /= (float)C;
  float inv = rsqrtf(var + 1e-5f);
  float* orow = out + m * C;
  for (int c = 0; c < C; ++c) orow[c] = (xr[c] - mu) * inv * g[c] + bta[c];
}

// depthwise 3x3 SAME + bias + SiLU; input = first d channels of xz (B,H,W,2d)
__global__ void dwconv3x3_silu(const float* __restrict__ xz, const float* __restrict__ w,
                               const float* __restrict__ bias, float* __restrict__ out,
                               int Bsz, int H, int W, int d) {
  long total = (long)Bsz * H * W * d;
  long i = (long)blockIdx.x * blockDim.x + threadIdx.x;
  if (i >= total) return;
  int c = (int)(i % d); long t = i / d;
  int x = (int)(t % W); t /= W;
  int y = (int)(t % H); int b = (int)(t / H);
  float acc = bias[c];
  for (int dy = -1; dy <= 1; ++dy)
    for (int dx = -1; dx <= 1; ++dx) {
      int yy = y + dy, xx = x + dx;
      if (yy >= 0 && yy < H && xx >= 0 && xx < W)
        acc += xz[(((long)b * H + yy) * W + xx) * (2L * d) + c] *
               w[c * 9 + (dy + 1) * 3 + (dx + 1)];
    }
  out[i] = acc / (1.f + __expf(-acc));  // SiLU
}

// build xs (B,4,d,L) from NHWC conv output: k0 hw, k1 wh, k2 rev-hw, k3 rev-wh
__global__ void build_xs(const float* __restrict__ xc, float* __restrict__ xs,
                         int Bsz, int H, int W, int d) {
  long L = (long)H * W;
  long total = (long)Bsz * 4 * d * L;
  long i = (long)blockIdx.x * blockDim.x + threadIdx.x;
  if (i >= total) return;
  int l = (int)(i % L); long t = i / L;
  int dch = (int)(t % d); t /= d;
  int k = (int)(t % 4); int b = (int)(t / 4);
  int seq = (k >= 2) ? (int)L - 1 - l : l;
  int h, w;
  if ((k & 1) == 0) { h = seq / W; w = seq % W; }
  else              { w = seq / H; h = seq % H; }
  xs[i] = xc[(((long)b * H + h) * W + w) * d + dch];
}

// ---------------------------------------------------------------------------
// selective scan, 16-wide over state dim: each 16-lane group owns one
// (b,k,dch) channel, one of the 16 states per lane. Per-step output
// reduction via __shfl_xor butterfly (wave32 crossbar). ys overwrites dts.
// ---------------------------------------------------------------------------
__global__ void selective_scan_kernel(const float* __restrict__ xs, float* __restrict__ dts_ys,
                                      const float* __restrict__ xdbl,
                                      const float* __restrict__ A_logs,
                                      const float* __restrict__ Ds,
                                      int Bsz, int d, int L, int RC, int R) {
  int Kd = 4 * d;
  long gid = (long)blockIdx.x * blockDim.x + threadIdx.x;
  int n = (int)(gid & 15);
  long ch = gid >> 4;  // channel index over (b,k,dch)
  if (ch >= (long)Bsz * Kd) return;
  int b = (int)(ch / Kd);
  int kd = (int)(ch % Kd);
  int k = kd / d;
  const float* x_p = xs + ch * L;
  float* dt_p = dts_ys + ch * L;
  const float* Brow = xdbl + (((long)b * 4 + k) * RC + R + n) * (long)L;
  const float* Crow = Brow + 16L * L;
  float Ac = -__expf(A_logs[(long)kd * 16 + n]);
  float Dv = Ds[kd];
  float h = 0.f;
  for (int l = 0; l < L; ++l) {
    float dt = dt_p[l];
    float xv = x_p[l];
    h = h * __expf(dt * Ac) + (dt * xv) * Brow[l];
    float y = h * Crow[l];
    y += __shfl_xor(y, 1, 32);
    y += __shfl_xor(y, 2, 32);
    y += __shfl_xor(y, 4, 32);
    y += __shfl_xor(y, 8, 32);
    if (n == 0) dt_p[l] = y + Dv * xv;
  }
}

// combine 4 direction outputs (B,4,d,L) -> (M, d) with m = b*L + (h*W+w)
__global__ void combine_dirs(const float* __restrict__ ys, float* __restrict__ y,
                             int Bsz, int H, int W, int d) {
  long L = (long)H * W;
  long total = (long)Bsz * L * d;
  long i = (long)blockIdx.x * blockDim.x + threadIdx.x;
  if (i >= total) return;
  int dch = (int)(i % d); long t = i / d;
  int l = (int)(t % L); int b = (int)(t / L);
  int h = l / W, w = l % W;
  int lwh = w * H + h;
  long base = ((long)b * 4) * d;
  const float* y0 = ys + (base + 0 * d + dch) * L;
  const float* y1 = ys + (base + 1L * d + dch) * L;
  const float* y2 = ys + (base + 2L * d + dch) * L;
  const float* y3 = ys + (base + 3L * d + dch) * L;
  y[((long)b * L + l) * d + dch] =
      y0[l] + y2[(long)L - 1 - l] + y1[lwh] + y3[(long)L - 1 - lwh];
}

// out_norm layernorm over d + multiply by silu(z); in-place on y; z = xz[:, d:2d]
__global__ void outnorm_gate(float* __restrict__ y, const float* __restrict__ xz,
                             const float* __restrict__ g, const float* __restrict__ bta,
                             long M, int d) {
  long m = (long)blockIdx.x * blockDim.x + threadIdx.x;
  if (m >= M) return;
  float* yr = y + m * d;
  const float* zr = xz + m * (2L * d) + d;
  float mu = 0.f;
  for (int c = 0; c < d; ++c) mu += yr[c];
  mu /= (float)d;
  float var = 0.f;
  for (int c = 0; c < d; ++c) { float dv = yr[c] - mu; var += dv * dv; }
  var /= (float)d;
  float inv = rsqrtf(var + 1e-5f);
  for (int c = 0; c < d; ++c) {
    float v = (yr[c] - mu) * inv * g[c] + bta[c];
    float z = zr[c];
    yr[c] = v * (z / (1.f + __expf(-z)));
  }
}

// x = x*skip[c] + add   (in-place)
__global__ void residual_scale_add(float* __restrict__ x, const float* __restrict__ addv,
                                   const float* __restrict__ skip, long M, int C) {
  long total = M * C;
  long i = (long)blockIdx.x * blockDim.x + threadIdx.x;
  if (i >= total) return;
  int c = (int)(i % C);
  x[i] = x[i] * skip[c] + addv[i];
}

// dense NHWC conv3x3 SAME; ACT: 0 none, 1 exact GELU
template <int ACT>
__global__ void conv3x3(const float* __restrict__ x, const float* __restrict__ w,
                        const float* __restrict__ bias, float* __restrict__ out,
                        int Bsz, int H, int W, int Cin, int Cout) {
  long total = (long)Bsz * H * W * Cout;
  long i = (long)blockIdx.x * blockDim.x + threadIdx.x;
  if (i >= total) return;
  int co = (int)(i % Cout); long t = i / Cout;
  int xx0 = (int)(t % W); t /= W;
  int yy0 = (int)(t % H); int b = (int)(t / H);
  float acc = bias[co];
  for (int dy = -1; dy <= 1; ++dy)
    for (int dx = -1; dx <= 1; ++dx) {
      int yy = yy0 + dy, xx = xx0 + dx;
      if (yy < 0 || yy >= H || xx < 0 || xx >= W) continue;
      const float* xp = x + (((long)b * H + yy) * W + xx) * Cin;
      const float* wp = w + (((long)co * Cin) * 3 + (dy + 1)) * 3 + (dx + 1);
      for (int ci = 0; ci < Cin; ++ci) acc += xp[ci] * wp[(long)ci * 9];
    }
  if (ACT == 1) acc = 0.5f * acc * (1.f + erff(acc * 0.70710678118f));
  out[i] = acc;
}

__global__ void pool_mean(const float* __restrict__ x, float* __restrict__ out,
                          int Bsz, int L, int C) {
  long i = (long)blockIdx.x * blockDim.x + threadIdx.x;
  if (i >= (long)Bsz * C) return;
  int c = (int)(i % C); int b = (int)(i / C);
  float s = 0.f;
  for (int l = 0; l < L; ++l) s += x[((long)b * L + l) * C + c];
  out[i] = s / (float)L;
}

// out[b,o] = act(bias[o] + sum_i in[b,i]*W[o*I+i]); act 0=relu 1=sigmoid
__global__ void ca_fc(const float* __restrict__ in, const float* __restrict__ W,
                      const float* __restrict__ bias, float* __restrict__ out,
                      int Bsz, int I, int O, int act) {
  long i = (long)blockIdx.x * blockDim.x + threadIdx.x;
  if (i >= (long)Bsz * O) return;
  int o = (int)(i % O); int b = (int)(i / O);
  float s = bias[o];
  for (int j = 0; j < I; ++j) s += in[(long)b * I + j] * W[(long)o * I + j];
  out[i] = (act == 0) ? fmaxf(s, 0.f) : 1.f / (1.f + __expf(-s));
}

// x = x*skip2[c] + c2[m,c]*a2[b,c]
__global__ void cab_residual(float* __restrict__ x, const float* __restrict__ c2,
                             const float* __restrict__ a2, const float* __restrict__ skip,
                             int Bsz, int L, int C) {
  long total = (long)Bsz * L * C;
  long i = (long)blockIdx.x * blockDim.x + threadIdx.x;
  if (i >= total) return;
  int c = (int)(i % C);
  int b = (int)(i / ((long)L * C));
  x[i] = x[i] * skip[c] + c2[i] * a2[(long)b * C + c];
}

// bilinear 2x upsample, align_corners=False (half-pixel), NHWC
__global__ void upsample_bilinear2x(const float* __restrict__ in, float* __restrict__ out,
                                    int Bsz, int H, int W, int C) {
  int H2 = 2 * H, W2 = 2 * W;
  long total = (long)Bsz * H2 * W2 * C;
  long i = (long)blockIdx.x * blockDim.x + threadIdx.x;
  if (i >= total) return;
  int c = (int)(i % C); long t = i / C;
  int x2 = (int)(t % W2); t /= W2;
  int y2 = (int)(t % H2); int b = (int)(t / H2);
  float sy = (y2 + 0.5f) * 0.5f - 0.5f;
  float sx = (x2 + 0.5f) * 0.5f - 0.5f;
  float fy = floorf(sy), fx = floorf(sx);
  float wy = sy - fy, wx = sx - fx;
  int y0 = (int)fy, x0 = (int)fx;
  int y0c = min(max(y0, 0), H - 1), y1c = min(max(y0 + 1, 0), H - 1);
  int x0c = min(max(x0, 0), W - 1), x1c = min(max(x0 + 1, 0), W - 1);
  const float* base = in + ((long)b * H) * W * C + c;
  float v00 = base[((long)y0c * W + x0c) * C];
  float v01 = base[((long)y0c * W + x1c) * C];
  float v10 = base[((long)y1c * W + x0c) * C];
  float v11 = base[((long)y1c * W + x1c) * C];
  out[i] = (1.f - wy) * ((1.f - wx) * v00 + wx * v01) + wy * ((1.f - wx) * v10 + wx * v11);
}

// x[m,c] += skip_nchw[b,c,l]
__global__ void add_skip_nchw(float* __restrict__ x, const float* __restrict__ skip,
                              int Bsz, int C, int L) {
  long total = (long)Bsz * L * C;
  long i = (long)blockIdx.x * blockDim.x + threadIdx.x;
  if (i >= total) return;
  int c = (int)(i % C); long t = i / C;
  int l = (int)(t % L); int b = (int)(t / L);
  x[i] += skip[((long)b * C + c) * L + l];
}

__global__ void nhwc_to_nchw(const float* __restrict__ in, float* __restrict__ out,
                             int Bsz, int H, int W, int C) {
  long total = (long)Bsz * C * H * W;
  long i = (long)blockIdx.x * blockDim.x + threadIdx.x;
  if (i >= total) return;
  int x = (int)(i % W); long t = i / W;
  int y = (int)(t % H); t /= H;
  int c = (int)(t % C); int b = (int)(t / C);
  out[i] = in[(((long)b * H + y) * W + x) * C + c];
}

// ---------------------------------------------------------------------------
// Host-side orchestration
// ---------------------------------------------------------------------------
struct BlockParams {
  const float *ln1_g, *ln1_b, *ln2_g, *ln2_b, *skip1, *skip2;
  const float *in_proj_w, *conv_w, *conv_b, *x_proj_w, *dt_projs_w, *dt_projs_b;
  const float *A_logs, *Ds, *out_norm_g, *out_norm_b, *out_proj_w;
  const float *c1_w, *c1_b, *c2_w, *c2_b, *ca1_w, *ca1_b, *ca2_w, *ca2_b;
};

static void launch_gemm(const float* A, const float* B, float* C, const float* bias,
                        int M, int N, int K, long am, long ak, long bk, long bn,
                        long cm, long cn, int act, hipStream_t s) {
  long tiles = cdivl(M, 16) * cdivl(N, 16);
  int blocks = (int)cdivl(tiles, 8);
  bool a1 = (ak == 1), b1 = (bk == 1);
  if (act == 0) {
    if (a1 && b1)
      gemm_wmma_f32<0, 1, 1><<<blocks, 256, 0, s>>>(A, B, C, bias, M, N, K, am, ak, bk, bn, cm, cn);
    else if (b1)
      gemm_wmma_f32<0, 0, 1><<<blocks, 256, 0, s>>>(A, B, C, bias, M, N, K, am, ak, bk, bn, cm, cn);
    else
      gemm_wmma_f32<0, 0, 0><<<blocks, 256, 0, s>>>(A, B, C, bias, M, N, K, am, ak, bk, bn, cm, cn);
  } else {
    if (a1 && b1)
      gemm_wmma_f32<1, 1, 1><<<blocks, 256, 0, s>>>(A, B, C, bias, M, N, K, am, ak, bk, bn, cm, cn);
    else if (b1)
      gemm_wmma_f32<1, 0, 1><<<blocks, 256, 0, s>>>(A, B, C, bias, M, N, K, am, ak, bk, bn, cm, cn);
    else
      gemm_wmma_f32<1, 0, 0><<<blocks, 256, 0, s>>>(A, B, C, bias, M, N, K, am, ak, bk, bn, cm, cn);
  }
}

static void run_vss_block(float* xio, const BlockParams& P, int Bsz, int H, int W, int C,
                          float* scr, hipStream_t s) {
  const int TB = 256;
  int d = 2 * C;
  int R = (C + 15) / 16; if (R < 1) R = 1;
  int RC = R + 32;
  long L = (long)H * W;
  long M = (long)Bsz * L;
  int Cc = C / 3; if (Cc < 1) Cc = 1;
  int Csq = C / 30; if (Csq < 1) Csq = 1;

  float* t_ln = scr;                           // (M, C)
  float* t_xz = t_ln + M * C;                  // (M, 2d)
  float* t_cv = t_xz + M * 2L * d;             // (M, d)   NHWC conv out
  float* t_xs = t_cv + M * d;                  // (B,4,d,L)
  float* t_xd = t_xs + M * 4L * d;             // (B,4,RC,L)
  float* t_dt = t_xd + (long)Bsz * 4 * RC * L; // (B,4,d,L) dts -> ys in-place
  float* t_y  = t_dt + M * 4L * d;             // (M, d)
  float* t_ss = t_y + M * d;                   // (M, C)
  float* t_c1 = t_ss + M * C;                  // (M, Cc)
  float* t_c2 = t_c1 + M * Cc;                 // (M, C)
  float* t_pl = t_c2 + M * C;                  // (B, C)
  float* t_a1 = t_pl + (long)Bsz * C;          // (B, Csq)
  float* t_a2 = t_a1 + (long)Bsz * Csq;        // (B, C)

  layernorm_rows<<<(int)cdivl(M, TB), TB, 0, s>>>(xio, P.ln1_g, P.ln1_b, t_ln, M, C);
  launch_gemm(t_ln, P.in_proj_w, t_xz, nullptr, (int)M, 2 * d, C, C, 1, 1, C, 2 * d, 1, 0, s);
  dwconv3x3_silu<<<(int)cdivl(M * d, TB), TB, 0, s>>>(t_xz, P.conv_w, P.conv_b, t_cv, Bsz, H, W, d);
  build_xs<<<(int)cdivl(M * 4 * d, TB), TB, 0, s>>>(t_cv, t_xs, Bsz, H, W, d);

  for (int b = 0; b < Bsz; ++b)
    for (int k = 0; k < 4; ++k) {
      const float* xsbk = t_xs + (((long)b * 4 + k) * d) * L;
      float* xdbk = t_xd + (((long)b * 4 + k) * RC) * L;
      // x_dbl[l, c] = sum_dd xs[dd, l] * x_proj_w[k][c, dd]
      launch_gemm(xsbk, P.x_proj_w + (long)k * RC * d, xdbk, nullptr,
                  (int)L, RC, d, 1, L, 1, d, 1, L, 0, s);
      // dts[l, dch] = softplus(sum_r xdbl[r, l] * dt_projs_w[k][dch, r] + b[dch])
      float* dtbk = t_dt + (((long)b * 4 + k) * d) * L;
      launch_gemm(xdbk, P.dt_projs_w + (long)k * d * R, dtbk, P.dt_projs_b + (long)k * d,
                  (int)L, d, R, 1, L, 1, R, 1, L, 1, s);
    }

  long nscan = (long)Bsz * 4 * d * 16;  // 16 lanes per channel
  selective_scan_kernel<<<(int)cdivl(nscan, 256), 256, 0, s>>>(
      t_xs, t_dt, t_xd, P.A_logs, P.Ds, Bsz, d, (int)L, RC, R);

  combine_dirs<<<(int)cdivl(M * d, TB), TB, 0, s>>>(t_dt, t_y, Bsz, H, W, d);
  outnorm_gate<<<(int)cdivl(M, TB), TB, 0, s>>>(t_y, t_xz, P.out_norm_g, P.out_norm_b, M, d);
  launch_gemm(t_y, P.out_proj_w, t_ss, nullptr, (int)M, C, d, d, 1, 1, d, C, 1, 0, s);
  residual_scale_add<<<(int)cdivl(M * C, TB), TB, 0, s>>>(xio, t_ss, P.skip1, M, C);

  // CAB branch
  layernorm_rows<<<(int)cdivl(M, TB), TB, 0, s>>>(xio, P.ln2_g, P.ln2_b, t_ln, M, C);
  conv3x3<1><<<(int)cdivl(M * Cc, TB), TB, 0, s>>>(t_ln, P.c1_w, P.c1_b, t_c1, Bsz, H, W, C, Cc);
  conv3x3<0><<<(int)cdivl(M * C, TB), TB, 0, s>>>(t_c1, P.c2_w, P.c2_b, t_c2, Bsz, H, W, Cc, C);
  pool_mean<<<(int)cdivl((long)Bsz * C, TB), TB, 0, s>>>(t_c2, t_pl, Bsz, (int)L, C);
  ca_fc<<<(int)cdivl((long)Bsz * Csq, 64), 64, 0, s>>>(t_pl, P.ca1_w, P.ca1_b, t_a1, Bsz, C, Csq, 0);
  ca_fc<<<(int)cdivl((long)Bsz * C, 64), 64, 0, s>>>(t_a1, P.ca2_w, P.ca2_b, t_a2, Bsz, Csq, C, 1);
  cab_residual<<<(int)cdivl(M * C, TB), TB, 0, s>>>(xio, t_c2, t_a2, P.skip2, Bsz, (int)L, C);
}

static BlockParams load_block(void* const* d_in, int base) {
  BlockParams P;
  const float** f = (const float**)&P;
  for (int j = 0; j < 25; ++j) f[j] = (const float*)d_in[base + j];
  return P;
}

extern "C" void kernel_launch(void* const* d_in, const int* in_sizes, int n_in,
                              void* d_out, int out_size, void* d_ws, size_t ws_size,
                              hipStream_t stream) {
  (void)in_sizes; (void)n_in; (void)out_size; (void)ws_size;
  const int TB = 256;
  const float* x      = (const float*)d_in[0];   // (2,64,64,64) NCHW
  const float* prompt = (const float*)d_in[1];   // (2,32,64,64) NCHW
  const float* skip   = (const float*)d_in[2];   // (2,32,128,128) NCHW
  BlockParams fuseP = load_block(d_in, 3);       // 3..27
  const float* reduce_w = (const float*)d_in[28];  // (64,96)
  const float* up_w     = (const float*)d_in[29];  // (32,64)
  BlockParams refP = load_block(d_in, 30);       // 30..54

  float* ws = (float*)d_ws;
  float* xin_f = ws;                    // (2*4096, 96) NHWC
  float* t_red = xin_f + 8192L * 96;    // (2*4096, 64)
  float* x_ref = t_red + 8192L * 64;    // (2*16384, 32) up out / refine io
  float* t_up  = x_ref + 32768L * 32;   // (2*16384, 64) upsampled
  float* scr   = t_up + 32768L * 64;    // stage scratch (shared)

  // fuse stage: concat -> VSS(C=96) on 64x64
  concat_nchw_to_nhwc<<<(int)cdivl(8192L * 96, TB), TB, 0, stream>>>(x, prompt, xin_f, 2, 64, 64);
  run_vss_block(xin_f, fuseP, 2, 64, 64, 96, scr, stream);

  // 1x1 reduce 96->64
  launch_gemm(xin_f, reduce_w, t_red, nullptr, 8192, 64, 96, 96, 1, 1, 96, 64, 1, 0, stream);
  // bilinear x2 -> (2,128,128,64)
  upsample_bilinear2x<<<(int)cdivl(32768L * 64, TB), TB, 0, stream>>>(t_red, t_up, 2, 64, 64, 64);
  // 1x1 up 64->32, then + skip
  launch_gemm(t_up, up_w, x_ref, nullptr, 32768, 32, 64, 64, 1, 1, 64, 32, 1, 0, stream);
  add_skip_nchw<<<(int)cdivl(32768L * 32, TB), TB, 0, stream>>>(x_ref, skip, 2, 32, 16384);

  // refine stage: VSS(C=32) on 128x128
  run_vss_block(x_ref, refP, 2, 128, 128, 32, scr, stream);

  // NHWC -> NCHW output
  nhwc_to_nchw<<<(int)cdivl(32768L * 32, TB), TB, 0, stream>>>(x_ref, (float*)d_out, 2, 128, 128, 32);
}